// TopKSparseAutoencoder_4071628997267
// MI455X (gfx1250) — compile-verified
//
#include <hip/hip_runtime.h>

typedef float v2f __attribute__((ext_vector_type(2)));
typedef float v8f __attribute__((ext_vector_type(8)));

#define BATCH 8192
#define IN_DIM 4096
#define HID_DIM 16384

#define BM 128
#define BN 128
#define KC 16
#define SK 20   // padded LDS row stride (floats): 80B rows -> 16B aligned, conflict-free frags

// Issue one 16-byte async global->LDS copy (ASYNCcnt tracked).
__device__ __forceinline__ void async_ld16(const float* gptr, const float* lptr) {
    const unsigned           lds = (unsigned)(uintptr_t)lptr;   // addr[31:0] = LDS offset
    const unsigned long long ga  = (unsigned long long)(uintptr_t)gptr;
    asm volatile("global_load_async_to_lds_b128 %0, %1, off"
                 :: "v"(lds), "v"(ga) : "memory");
}

__device__ __forceinline__ void wait_async0() {
    asm volatile("s_wait_asynccnt 0x0" ::: "memory");
}

// C[M,N] (ld = N) = op( A[M,K] * Bt[N,K]^T + bias ), op = ReLU if do_relu.
// A rows and Bt rows are both contiguous along K (memory layout of x / W_enc / W_dec).
__global__ __launch_bounds__(256)
void gemm_nt_wmma(const float* __restrict__ A,
                  const float* __restrict__ Bt,
                  const float* __restrict__ bias,
                  float* __restrict__ C,
                  int K, int N, int do_relu)
{
    __shared__ float lA[2][BM * SK];
    __shared__ float lB[2][BN * SK];

    const int tid  = threadIdx.x;
    const int lane = tid & 31;
    const int wave = tid >> 5;
    const int wm   = wave >> 2;   // 0..1 : 64-row slab of the 128x128 tile
    const int wn   = wave & 3;    // 0..3 : 32-col slab

    const int mBlk = blockIdx.y * BM;
    const int nBlk = blockIdx.x * BN;

    v8f acc[4][2] = {};

    // Staging assignment: 256 threads x 16B, 2 passes cover a 128x16 tile.
    const int lrow = tid >> 2;         // 0..63
    const int lcol = (tid & 3) << 2;   // 0,4,8,12

    const int nChunks = K / KC;

    // Prologue: stage chunk 0 into buffer 0.
    #pragma unroll
    for (int r = 0; r < BM; r += 64) {
        const int row = r + lrow;
        async_ld16(&A [(size_t)(mBlk + row) * K + lcol], &lA[0][row * SK + lcol]);
        async_ld16(&Bt[(size_t)(nBlk + row) * K + lcol], &lB[0][row * SK + lcol]);
    }

    for (int c = 0; c < nChunks; ++c) {
        const int buf = c & 1;

        wait_async0();          // this wave's copies into lA/lB[buf] done
        __syncthreads();        // everyone's copies done

        // Stage chunk c+1 into the other buffer while we compute on buf.
        if (c + 1 < nChunks) {
            const int k0n = (c + 1) * KC;
            #pragma unroll
            for (int r = 0; r < BM; r += 64) {
                const int row = r + lrow;
                async_ld16(&A [(size_t)(mBlk + row) * K + k0n + lcol],
                           &lA[buf ^ 1][row * SK + lcol]);
                async_ld16(&Bt[(size_t)(nBlk + row) * K + k0n + lcol],
                           &lB[buf ^ 1][row * SK + lcol]);
            }
        }

        #pragma unroll
        for (int kk = 0; kk < KC; kk += 4) {
            // f32 WMMA 16x16x4 fragment layout:
            // lanes 0-15 hold K = kk,kk+1 ; lanes 16-31 hold K = kk+2,kk+3
            const int colk = kk + ((lane >> 4) << 1);
            v2f av[4], bv[2];
            #pragma unroll
            for (int i = 0; i < 4; ++i) {
                const int m = wm * 64 + i * 16 + (lane & 15);
                av[i] = *reinterpret_cast<const v2f*>(&lA[buf][m * SK + colk]);
            }
            #pragma unroll
            for (int j = 0; j < 2; ++j) {
                const int n = wn * 32 + j * 16 + (lane & 15);
                bv[j] = *reinterpret_cast<const v2f*>(&lB[buf][n * SK + colk]);
            }
            #pragma unroll
            for (int i = 0; i < 4; ++i)
                #pragma unroll
                for (int j = 0; j < 2; ++j)
                    acc[i][j] = __builtin_amdgcn_wmma_f32_16x16x4_f32(
                        false, av[i], false, bv[j],
                        (short)0, acc[i][j], false, false);
        }

        __syncthreads();        // buf is free to be overwritten two chunks later
    }

    // Epilogue: C/D layout -> VGPR e holds row (e) for lanes 0-15, row (e+8) for lanes 16-31.
    const int mh = (lane >> 4) << 3;
    #pragma unroll
    for (int i = 0; i < 4; ++i) {
        #pragma unroll
        for (int j = 0; j < 2; ++j) {
            const int n  = nBlk + wn * 32 + j * 16 + (lane & 15);
            const float bb = bias ? bias[n] : 0.0f;
            #pragma unroll
            for (int e = 0; e < 8; ++e) {
                const int m = mBlk + wm * 64 + i * 16 + mh + e;
                float v = acc[i][j][e] + bb;
                if (do_relu) v = fmaxf(v, 0.0f);
                C[(size_t)m * N + n] = v;
            }
        }
    }
}

// One block per batch row. ReLU output >= 0, so float order == uint bit order:
// binary-search the k-th-largest bit pattern, then zero everything below it (in place).
__global__ __launch_bounds__(256)
void topk_mask(float* __restrict__ feat, const int* __restrict__ kptr)
{
    const int row = blockIdx.x;
    const int tid = threadIdx.x;
    const int k   = *kptr;
    float* p = feat + (size_t)row * HID_DIM;

    float v[64];
    #pragma unroll
    for (int i = 0; i < 64; ++i)
        v[i] = p[i * 256 + tid];

    __shared__ int sh;
    unsigned lo = 0u, hi = 0x7F800000u;   // [0, +inf)
    while (hi - lo > 1u) {                 // uniform across block
        const unsigned mid = lo + ((hi - lo) >> 1);
        int c = 0;
        #pragma unroll
        for (int i = 0; i < 64; ++i)
            c += (__float_as_uint(v[i]) >= mid) ? 1 : 0;
        __syncthreads();
        if (tid == 0) sh = 0;
        __syncthreads();
        atomicAdd(&sh, c);
        __syncthreads();
        const int total = sh;
        if (total >= k) lo = mid; else hi = mid;
        __syncthreads();
    }

    #pragma unroll
    for (int i = 0; i < 64; ++i) {
        const float out = (__float_as_uint(v[i]) >= lo) ? v[i] : 0.0f;
        p[i * 256 + tid] = out;
    }
}

extern "C" void kernel_launch(void* const* d_in, const int* in_sizes, int n_in,
                              void* d_out, int out_size, void* d_ws, size_t ws_size,
                              hipStream_t stream) {
    const float* x     = (const float*)d_in[0];   // [B, D]
    const float* W_enc = (const float*)d_in[1];   // [H, D]
    const float* b_enc = (const float*)d_in[2];   // [H]
    const float* W_dec = (const float*)d_in[3];   // [D, H]
    const int*   kptr  = (const int*)d_in[4];     // scalar k

    float* feat  = (float*)d_out;                          // sparse_features [B, H]
    float* recon = feat + (size_t)BATCH * HID_DIM;         // reconstruction  [B, D]

    dim3 blk(256);

    // Encoder: feat = relu(x @ W_enc^T + b_enc)   (K = IN_DIM, N = HID_DIM)
    gemm_nt_wmma<<<dim3(HID_DIM / BN, BATCH / BM), blk, 0, stream>>>(
        x, W_enc, b_enc, feat, IN_DIM, HID_DIM, 1);

    // TopK mask in place on feat.
    topk_mask<<<dim3(BATCH), blk, 0, stream>>>(feat, kptr);

    // Decoder: recon = feat @ W_dec^T             (K = HID_DIM, N = IN_DIM)
    gemm_nt_wmma<<<dim3(IN_DIM / BN, BATCH / BM), blk, 0, stream>>>(
        feat, W_dec, nullptr, recon, HID_DIM, IN_DIM, 0);
}